// CBowHSModel_75153337745591
// MI455X (gfx1250) — compile-verified
//
#include <hip/hip_runtime.h>
#include <math.h>

// CDNA5 wave32 WMMA: D(16x16,f32) = A(16x4,f32) x B(4x16,f32) + C
typedef __attribute__((ext_vector_type(2))) float v2f;
typedef __attribute__((ext_vector_type(8))) float v8f;

#define EMBED 128
#define WAVES 4                 // waves per block
#define TILE  16                // samples per wave (one WMMA M dimension)
#define PAD   132               // padded LDS row stride (floats): (132%64)=4 -> conflict-free transposed reads
#define SAMPLES_PER_BLOCK (WAVES * TILE)

__device__ __forceinline__ float log_sigmoid_f(float x) {
    // stable: log_sigmoid(x) = min(x,0) - log1p(exp(-|x|))
    return fminf(x, 0.0f) - log1pf(expf(-fabsf(x)));
}

__global__ __launch_bounds__(WAVES * 32)
void cbow_hs_loss_kernel(const float* __restrict__ u_emb,
                         const float* __restrict__ w_emb,
                         const int*   __restrict__ pos_u,
                         const int*   __restrict__ pos_w,
                         const int*   __restrict__ neg_u,
                         const int*   __restrict__ neg_w,
                         float*       __restrict__ partials,
                         int N, int C)
{
    __shared__ float ldsS[WAVES][TILE][PAD];   // context-sum tile per wave
    __shared__ float ldsT[WAVES][TILE][PAD];   // target tile per wave
    __shared__ float blockAcc[WAVES];

    const int tid  = threadIdx.x;
    const int wave = tid >> 5;
    const int lane = tid & 31;
    const int sampleBase = blockIdx.x * SAMPLES_PER_BLOCK + wave * TILE;

    // WMMA f32 16x16x4 per-lane layout (ISA 7.12.2):
    // lanes 0-15 hold K={0,1}, lanes 16-31 hold K={2,3} of each K4 chunk.
    const int half = lane >> 4;
    const int mrow = lane & 15;

    float waveSum = 0.0f;

    #pragma unroll
    for (int pass = 0; pass < 2; ++pass) {
        const int*  ctxIdx = pass ? neg_u : pos_u;
        const int*  tgtIdx = pass ? neg_w : pos_w;
        const float sign   = pass ? -1.0f : 1.0f;

        // ---- Phase 1: coalesced gather + context sum-pool into LDS tiles ----
        for (int s = 0; s < TILE; ++s) {
            int n  = sampleBase + s;
            int nc = (n < N) ? n : (N - 1);
            const long long ibase = (long long)nc * C;
            float4 acc = make_float4(0.f, 0.f, 0.f, 0.f);
            for (int c = 0; c < C; ++c) {
                int idx = ctxIdx[ibase + c];                       // wave-uniform -> SMEM load
                const float4* row = (const float4*)(u_emb + (long long)idx * EMBED);
                float4 r = row[lane];                              // 512B fully coalesced
                acc.x += r.x; acc.y += r.y; acc.z += r.z; acc.w += r.w;
            }
            *(float4*)&ldsS[wave][s][lane * 4] = acc;
            int tw = tgtIdx[nc];                                   // wave-uniform
            const float4* trow = (const float4*)(w_emb + (long long)tw * EMBED);
            *(float4*)&ldsT[wave][s][lane * 4] = trow[lane];       // 512B fully coalesced
        }

        // order the cross-lane LDS store -> transposed load handoff (same wave)
        __asm__ volatile("s_wait_dscnt 0" ::: "memory");

        // ---- Phase 2: f32 WMMA dot products, D[m][m] = dot(S[m], T[m]) ----
        v8f c8 = {};
        const float* Srow = &ldsS[wave][mrow][0];
        const float* Trow = &ldsT[wave][mrow][0];
        #pragma unroll
        for (int k = 0; k < EMBED / 4; ++k) {
            int e = 4 * k + 2 * half;                  // conflict-free banks via PAD=132
            v2f a = *(const v2f*)(Srow + e);           // A[m][K] chunk
            v2f b = *(const v2f*)(Trow + e);           // B[K][n] = T[n][e+K] chunk
            c8 = __builtin_amdgcn_wmma_f32_16x16x4_f32(
                     false, a, false, b, (short)0, c8, false, false);
        }

        // ---- Extract diagonal D[m][m] (C/D layout: VGPR r -> M=r (lanes 0-15), M=r+8 (lanes 16-31)) ----
        const bool owns = (lane < 8) || (lane >= 24);
        const int  m    = (lane < 8) ? lane : (lane - 16);   // sample index within tile
        const int  comp = (lane < 8) ? lane : (lane - 24);   // VGPR component holding D[m][m]
        float sc = 0.0f;
        #pragma unroll
        for (int j = 0; j < 8; ++j) sc = (comp == j) ? c8[j] : sc;

        const bool valid = owns && (sampleBase + m < N);
        float contrib = valid ? -log_sigmoid_f(sign * sc) : 0.0f;

        // wave32 reduction
        #pragma unroll
        for (int off = 16; off > 0; off >>= 1)
            contrib += __shfl_xor(contrib, off, 32);
        waveSum += contrib;                                  // same value in all lanes
    }

    if (lane == 0) blockAcc[wave] = waveSum;
    __syncthreads();
    if (tid == 0) {
        float t = 0.0f;
        #pragma unroll
        for (int w = 0; w < WAVES; ++w) t += blockAcc[w];
        partials[blockIdx.x] = t;
    }
}

__global__ __launch_bounds__(256)
void reduce_partials_kernel(const float* __restrict__ partials, int n,
                            float* __restrict__ out)
{
    __shared__ float sm[256];
    float s = 0.0f;
    for (int i = threadIdx.x; i < n; i += 256) s += partials[i];   // fixed order: deterministic
    sm[threadIdx.x] = s;
    __syncthreads();
    for (int off = 128; off > 0; off >>= 1) {
        if ((int)threadIdx.x < off) sm[threadIdx.x] += sm[threadIdx.x + off];
        __syncthreads();
    }
    if (threadIdx.x == 0) out[0] = sm[0];
}

extern "C" void kernel_launch(void* const* d_in, const int* in_sizes, int n_in,
                              void* d_out, int out_size, void* d_ws, size_t ws_size,
                              hipStream_t stream) {
    const float* u_emb = (const float*)d_in[0];
    const float* w_emb = (const float*)d_in[1];
    const int*   pos_u = (const int*)d_in[2];
    const int*   pos_w = (const int*)d_in[3];
    const int*   neg_u = (const int*)d_in[4];
    const int*   neg_w = (const int*)d_in[5];

    const int N = in_sizes[3];                 // pos_w element count
    const int C = in_sizes[2] / N;             // context width (10)

    float* partials = (float*)d_ws;
    const int blocks = (N + SAMPLES_PER_BLOCK - 1) / SAMPLES_PER_BLOCK;

    cbow_hs_loss_kernel<<<blocks, WAVES * 32, 0, stream>>>(
        u_emb, w_emb, pos_u, pos_w, neg_u, neg_w, partials, N, C);

    reduce_partials_kernel<<<1, 256, 0, stream>>>(partials, blocks, (float*)d_out);
}